// Detection_47459388620825
// MI455X (gfx1250) — compile-verified
//
#include <hip/hip_runtime.h>
#include <math.h>

// ---------------- problem constants (match reference) ----------------
#define NUM_CLASSES 2
#define BATCH 32
#define CCH 7                       // reg(4) + obj(1) + cls(2)
#define CELL 8                      // 7 channels + count, 32B per cell

// per-scale grids
#define H0 60
#define W0 80
#define H1 30
#define W1 40
#define H2 15
#define W2 20
#define HW0 (H0 * W0)               // 4800
#define HW1 (H1 * W1)               // 1200
#define HW2 (H2 * W2)               // 300
#define A_TOT (HW0 + HW1 + HW2)     // 6300

#define SEG0 (BATCH * HW0)          // 153600
#define SEG1 (BATCH * HW1)          // 38400
#define SEG2 (BATCH * HW2)          // 9600
#define TOTAL_SEG (SEG0 + SEG1 + SEG2)  // 201600 cells * 32B = 6.45 MB (fits L2)

#define STRIDE0 3.0f
#define STRIDE1 6.0f
#define STRIDE2 12.0f

// ---------------- helpers ----------------
__device__ __forceinline__ void atomic_fadd_noret(float* p, float v) {
    // relaxed, agent scope -> hardware GLOBAL_ATOMIC_ADD_F32; return value
    // unused -> without-return form (STOREcnt), resolved at L2.
    (void)__hip_atomic_fetch_add(p, v, __ATOMIC_RELAXED, __HIP_MEMORY_SCOPE_AGENT);
}

// ---------------- kernel: scatter-sum one scale (defined FIRST so its asm
// shows in the disasm snippet — verify global_atomic_add_f32 lowering) -------
// sums layout per cell (8 f32): [reg.x, reg.y, reg.z, reg.w, obj, cls0, cls1, count]
__global__ __launch_bounds__(256) void det_scatter(
    const float*  __restrict__ cls,    // [n, 2]
    const float4* __restrict__ reg,    // [n, 4]
    const float*  __restrict__ obj,    // [n, 1]
    const float2* __restrict__ pos,    // [n, 2]
    const int*    __restrict__ batch,  // [n]
    float*        __restrict__ sums,   // [B*H*W, 8] for this scale
    int n, int W, int H, float stride)
{
    int i = blockIdx.x * blockDim.x + threadIdx.x;
    if (i >= n) return;

    // gfx1250 prefetch on the fattest streams; distance covers HBM latency.
    const int PF = 4096;
    if (i + PF < n) {
        __builtin_prefetch(&reg[i + PF], 0, 0);   // global_prefetch_b8
        __builtin_prefetch(&pos[i + PF], 0, 0);
    }

    float2 p = pos[i];                                 // global_load_b64
    int col = (int)(p.x / stride);
    int row = (int)(p.y / stride);
    col = min(max(col, 0), W - 1);
    row = min(max(row, 0), H - 1);
    int b = batch[i];                                  // global_load_b32

    float* cell = sums + (size_t)(b * (H * W) + row * W + col) * CELL;

    float4 r = reg[i];                                 // global_load_b128
    float  o = obj[i];                                 // global_load_b32
    float2 c = *(const float2*)(cls + 2 * (size_t)i);  // global_load_b64

    atomic_fadd_noret(cell + 0, r.x);
    atomic_fadd_noret(cell + 1, r.y);
    atomic_fadd_noret(cell + 2, r.z);
    atomic_fadd_noret(cell + 3, r.w);
    atomic_fadd_noret(cell + 4, o);
    atomic_fadd_noret(cell + 5, c.x);
    atomic_fadd_noret(cell + 6, c.y);
    atomic_fadd_noret(cell + 7, 1.0f);
}

// ---------------- kernel: zero the accumulator (L2-resident) ----------------
__global__ __launch_bounds__(256) void det_zero_ws(float4* __restrict__ ws, int n4) {
    int i = blockIdx.x * blockDim.x + threadIdx.x;
    if (i < n4) ws[i] = make_float4(0.f, 0.f, 0.f, 0.f);
}

// ---------------- kernel: mean + sigmoid + YOLOX decode ----------------
// out: [B, 6300, 7]; anchors concatenated scale0|scale1|scale2, row-major per scale.
__global__ __launch_bounds__(256) void det_decode(
    const float* __restrict__ sums,   // [TOTAL_SEG, 8]
    float*       __restrict__ out)
{
    int t = blockIdx.x * blockDim.x + threadIdx.x;
    if (t >= BATCH * A_TOT) return;

    int b = t / A_TOT;
    int a = t - b * A_TOT;

    int segBase, HW, Wd, idx;
    float stride;
    if (a < HW0)            { segBase = 0;           HW = HW0; Wd = W0; stride = STRIDE0; idx = a; }
    else if (a < HW0 + HW1) { segBase = SEG0;        HW = HW1; Wd = W1; stride = STRIDE1; idx = a - HW0; }
    else                    { segBase = SEG0 + SEG1; HW = HW2; Wd = W2; stride = STRIDE2; idx = a - HW0 - HW1; }

    const float* cell = sums + (size_t)(segBase + b * HW + idx) * CELL;
    float4 s0 = *(const float4*)(cell);       // reg sums       (b128)
    float4 s1 = *(const float4*)(cell + 4);   // obj, cls0, cls1, count

    float inv = 1.0f / fmaxf(s1.w, 1.0f);     // scatter_mean with max(count,1)
    float rx = s0.x * inv, ry = s0.y * inv;
    float rz = s0.z * inv, rw = s0.w * inv;
    float ob = s1.x * inv, c0 = s1.y * inv, c1 = s1.z * inv;

    int row = idx / Wd;
    int col = idx - row * Wd;

    float x = (rx + (float)col) * stride;
    float y = (ry + (float)row) * stride;
    float w = expf(fminf(rz, 10.0f)) * stride;
    float h = expf(fminf(rw, 10.0f)) * stride;
    ob = 1.0f / (1.0f + expf(-ob));
    c0 = 1.0f / (1.0f + expf(-c0));
    c1 = 1.0f / (1.0f + expf(-c1));

    float* o = out + (size_t)t * CCH;
    o[0] = x; o[1] = y; o[2] = w; o[3] = h;
    o[4] = ob; o[5] = c0; o[6] = c1;
}

// ---------------- launcher ----------------
extern "C" void kernel_launch(void* const* d_in, const int* in_sizes, int n_in,
                              void* d_out, int out_size, void* d_ws, size_t ws_size,
                              hipStream_t stream) {
    (void)n_in; (void)out_size; (void)ws_size;

    float* sums = (float*)d_ws;   // TOTAL_SEG * 8 f32 = 6.45 MB, L2-resident

    // zero accumulator
    {
        int n4 = TOTAL_SEG * CELL / 4;   // 403200 float4
        det_zero_ws<<<(n4 + 255) / 256, 256, 0, stream>>>((float4*)d_ws, n4);
    }

    // per-scale scatter: d_in order = cls, reg, obj, pos, batch per scale
    // n per scale taken from obj's element count (shape [n,1])
    {
        int n1 = in_sizes[2];
        det_scatter<<<(n1 + 255) / 256, 256, 0, stream>>>(
            (const float*)d_in[0], (const float4*)d_in[1], (const float*)d_in[2],
            (const float2*)d_in[3], (const int*)d_in[4],
            sums + 0, n1, W0, H0, STRIDE0);

        int n2 = in_sizes[7];
        det_scatter<<<(n2 + 255) / 256, 256, 0, stream>>>(
            (const float*)d_in[5], (const float4*)d_in[6], (const float*)d_in[7],
            (const float2*)d_in[8], (const int*)d_in[9],
            sums + (size_t)SEG0 * CELL, n2, W1, H1, STRIDE1);

        int n3 = in_sizes[12];
        det_scatter<<<(n3 + 255) / 256, 256, 0, stream>>>(
            (const float*)d_in[10], (const float4*)d_in[11], (const float*)d_in[12],
            (const float2*)d_in[13], (const int*)d_in[14],
            sums + (size_t)(SEG0 + SEG1) * CELL, n3, W2, H2, STRIDE2);
    }

    // decode to [B, 6300, 7]
    {
        int nt = BATCH * A_TOT;   // 201600
        det_decode<<<(nt + 255) / 256, 256, 0, stream>>>(sums, (float*)d_out);
    }
}